// SpatialEncoding_57354993271159
// MI455X (gfx1250) — compile-verified
//
#include <hip/hip_runtime.h>
#include <hip/hip_bf16.h>

typedef __attribute__((ext_vector_type(2))) float v2f;
typedef __attribute__((ext_vector_type(8))) float v8f;

#define NN 4096
#define DD 256
#define HH 8

// ---------------------------------------------------------------------------
// Kernel 1: per-row dot products xr = x@w_row, xc = x@w_col, and coord norms.
// One wave32 per row; butterfly reduce with shfl_xor.
// ---------------------------------------------------------------------------
__global__ void __launch_bounds__(128)
spatial_prep_kernel(const float* __restrict__ x,
                    const float* __restrict__ coords,
                    const float* __restrict__ wdist,   // 2*D+1 floats
                    float* __restrict__ xr,
                    float* __restrict__ xc,
                    float* __restrict__ norms)
{
    const int lane = threadIdx.x & 31;
    const int row  = blockIdx.x * 4 + (threadIdx.x >> 5);   // 1024 blocks * 4 waves
    const float* xrow = x + (size_t)row * DD;

    float sr = 0.0f, sc = 0.0f;
#pragma unroll
    for (int k = 0; k < DD / 32; ++k) {
        const int idx = lane + 32 * k;
        const float xv = xrow[idx];
        sr += xv * wdist[idx];            // w_row = wdist[0:D]
        sc += xv * wdist[DD + 1 + idx];   // w_col = wdist[D+1:2D+1]
    }
#pragma unroll
    for (int off = 16; off > 0; off >>= 1) {
        sr += __shfl_xor(sr, off, 32);
        sc += __shfl_xor(sc, off, 32);
    }
    if (lane == 0) {
        xr[row] = sr;
        xc[row] = sc;
        const float a = coords[row * 3 + 0];
        const float b = coords[row * 3 + 1];
        const float c = coords[row * 3 + 2];
        norms[row] = a * a + b * b + c * c;
    }
}

// ---------------------------------------------------------------------------
// Kernel 2: each wave32 computes a 16-row strip over 16 consecutive 16x16
// j-tiles (256 output columns). One V_WMMA_F32_16X16X4_F32 per tile produces
// the full squared-distance tile:  d2 = (norm_i + norm_j) + (-2*Ci) . Cj
// (A pre-scaled by -2, C seeded with norm_i + norm_j, K padded 3 -> 4).
// Then VALU: sqrt(|d2|), s = xr + wd*dist + xc, and the 8-gaussian exp2 loop
// with the 1/8 mean folded into the exponent (exp2(z - 3)).
// ---------------------------------------------------------------------------
__global__ void __launch_bounds__(256)
spatial_tile_kernel(const float* __restrict__ coords,
                    const float* __restrict__ means,
                    const float* __restrict__ stds,
                    const float* __restrict__ wdist,   // wdist[DD] = w_dist scalar
                    const float* __restrict__ xr,
                    const float* __restrict__ xc,
                    const float* __restrict__ norms,
                    float* __restrict__ out)
{
    const int lane   = threadIdx.x & 31;
    const int wave   = threadIdx.x >> 5;
    const int waveId = blockIdx.x * 8 + wave;   // 512 blocks * 8 waves = 4096
    const int tileRow = waveId >> 4;            // 0..255  -> i0 = tileRow*16
    const int jchunk  = waveId & 15;            // 0..15   -> 256-col chunk
    const int i0 = tileRow * 16;

    const int n  = lane & 15;    // column-in-tile for B/C/D layout
    const int hi = lane >> 4;    // upper half-wave selector

    // ---- A matrix (16x4 f32 layout): lanes 0-15 hold K=0,1; lanes 16-31 K=2,3
    // Row M = lane&15. Pre-scaled by -2; K=3 padded with zero.
    const float cax = coords[(size_t)(i0 + n) * 3 + 0];
    const float cay = coords[(size_t)(i0 + n) * 3 + 1];
    const float caz = coords[(size_t)(i0 + n) * 3 + 2];
    v2f a;
    a.x = -2.0f * (hi ? caz : cax);
    a.y = hi ? 0.0f : (-2.0f * cay);

    // ---- hoisted per-i-row values: C/D layout row m = r + 8*hi
    float xrv[8], niv[8];
#pragma unroll
    for (int r = 0; r < 8; ++r) {
        const int m = r + 8 * hi;
        xrv[r] = xr[i0 + m];
        niv[r] = norms[i0 + m];
    }

    const float wd = wdist[DD];
    float mh[HH], ch[HH];
#pragma unroll
    for (int h = 0; h < HH; ++h) {
        mh[h] = means[h];
        const float s = stds[h];
        // (1/8) * exp(t^2/(2 s^2)) = exp2(t^2 * log2(e)/(2 s^2) - 3)
        ch[h] = 0.72134752044448170f / (s * s);
    }

    for (int jj = 0; jj < 16; ++jj) {
        const int j0 = jchunk * 256 + jj * 16;
        const int jn = j0 + n;

        // ---- B matrix (4x16 f32): lanes 0-15 rows K=0,1; lanes 16-31 K=2,3
        const float cbx = coords[(size_t)jn * 3 + 0];
        const float cby = coords[(size_t)jn * 3 + 1];
        const float cbz = coords[(size_t)jn * 3 + 2];
        v2f b;
        b.x = hi ? cbz : cbx;
        b.y = hi ? 0.0f : cby;

        const float nj  = norms[jn];
        const float xcj = xc[jn];

        // ---- C seed = norm_i + norm_j, then one WMMA gives the d2 tile
        v8f c;
#pragma unroll
        for (int r = 0; r < 8; ++r) c[r] = niv[r] + nj;

        c = __builtin_amdgcn_wmma_f32_16x16x4_f32(
                /*neg_a=*/false, a, /*neg_b=*/false, b,
                /*c_mod=*/(short)0, c, /*reuse_a=*/false, /*reuse_b=*/false);

        // ---- per-element gaussian mixture
#pragma unroll
        for (int r = 0; r < 8; ++r) {
            // d2 >= 0 up to rounding; |.| is a free src modifier on v_sqrt_f32
            const float dist = __builtin_amdgcn_sqrtf(__builtin_fabsf(c[r]));
            const float s    = fmaf(wd, dist, xrv[r] + xcj);
            float acc = 0.0f;
#pragma unroll
            for (int h = 0; h < HH; ++h) {
                const float t  = s - mh[h];
                const float z2 = fmaf(t * t, ch[h], -3.0f);  // folds the /8 mean
                acc += __builtin_amdgcn_exp2f(z2);
            }
            const int m = r + 8 * hi;
            out[(size_t)(i0 + m) * NN + j0 + n] = acc;
        }
    }
}

extern "C" void kernel_launch(void* const* d_in, const int* in_sizes, int n_in,
                              void* d_out, int out_size, void* d_ws, size_t ws_size,
                              hipStream_t stream) {
    const float* x      = (const float*)d_in[0];   // (N, D)
    const float* coords = (const float*)d_in[1];   // (N, 3)
    const float* means  = (const float*)d_in[2];   // (H,)
    const float* stds   = (const float*)d_in[3];   // (H,)
    const float* wdist  = (const float*)d_in[4];   // (2D+1,)

    float* out   = (float*)d_out;
    float* xr    = (float*)d_ws;                   // N floats
    float* xc    = xr + NN;                        // N floats
    float* norms = xc + NN;                        // N floats  (48 KB total)

    // Kernel 1: 4096 rows, one wave per row, 4 waves per 128-thread block.
    spatial_prep_kernel<<<NN / 4, 128, 0, stream>>>(x, coords, wdist, xr, xc, norms);

    // Kernel 2: 4096 waves (256 i-tiles x 16 j-chunks), 8 waves per block.
    spatial_tile_kernel<<<512, 256, 0, stream>>>(coords, means, stds, wdist,
                                                 xr, xc, norms, out);
}